// KANLayer_89953795047906
// MI455X (gfx1250) — compile-verified
//
#include <hip/hip_runtime.h>

// Problem sizes (fixed by the reference).
#define B_SZ 16384
#define D_SZ 256
#define H_SZ 64
#define O_SZ 256

#define BM    64            // rows of x per workgroup
#define UPAD  260           // u-tile LDS row stride (floats): 260 % 64 == 4 -> conflict-free WMMA frag loads

// PWL tables (transposed [pos][D] so LDS bank = f(d) -> conflict-free divergent gathers):
//   BP: 64 x 256 sorted breakpoints (float)
//   SI: 65 x 256 per-segment (slope,intercept) pairs (float2)
#define BP_F   (64 * D_SZ)                        // floats
#define SI_F   (65 * D_SZ * 2)                    // floats (float2 interleaved)
#define TBL_F  (BP_F + SI_F)                      // 49,664 floats = 198,656 B
#define SMEM_PWL_F   (TBL_F + BM * UPAD)          // 66,304 floats = 265,216 B
#define SMEM_DIR_F   (BM * UPAD)                  //  66,560 B

// Async Global->LDS table staging (CDNA5 ASYNCcnt path). Set to 0 to fall back
// to register-staged copies.
#define KAN_ASYNC_COPY 1

typedef float v2f __attribute__((ext_vector_type(2)));
typedef float v8f __attribute__((ext_vector_type(8)));

// ---------------------------------------------------------------------------
// Precompute: per feature d, build the exact piecewise-linear form of
//   u(x) = sum_h relu(x*w1+b1)*w2 + b2
// A term with w1>0 activates when x crosses t=-b1/w1 upward; w1<0 deactivates.
// ---------------------------------------------------------------------------
__global__ __launch_bounds__(64) void kan_build_pwl(
    const float* __restrict__ W1, const float* __restrict__ b1,
    const float* __restrict__ W2, const float* __restrict__ b2,
    float* __restrict__ BP, float2* __restrict__ SI)
{
    const int d = blockIdx.x;     // 0..255
    const int h = threadIdx.x;    // 0..63

    __shared__ float t_s[H_SZ];
    __shared__ float bsl_s[H_SZ], bic_s[H_SZ];
    __shared__ float t_so[H_SZ], dsl_so[H_SZ], dic_so[H_SZ];
    __shared__ float seg_sl[H_SZ + 1], seg_ic[H_SZ + 1];

    const float w1 = W1[d * H_SZ + h];
    const float bb = b1[d * H_SZ + h];
    const float w2 = W2[d * H_SZ + h];
    const float s = w1 * w2;      // slope contribution while active
    const float c = bb * w2;      // intercept contribution while active

    float t, dsl, dic, bsl = 0.f, bic = 0.f;
    if (w1 > 0.f) {               // activates at t (x rising)
        t = -bb / w1; dsl = s;  dic = c;
    } else if (w1 < 0.f) {        // active at -inf, deactivates at t
        t = -bb / w1; dsl = -s; dic = -c; bsl = s; bic = c;
    } else {                      // constant term relu(b1)*w2
        t = __builtin_inff(); dsl = 0.f; dic = 0.f; bic = fmaxf(bb, 0.f) * w2;
    }

    t_s[h] = t; bsl_s[h] = bsl; bic_s[h] = bic;
    __syncthreads();

    // Stable rank sort (64 compares per thread), scatter into sorted order.
    int rank = 0;
    #pragma unroll 8
    for (int j = 0; j < H_SZ; ++j) {
        const float tj = t_s[j];
        rank += (tj < t) || (tj == t && j < h);
    }
    t_so[rank] = t; dsl_so[rank] = dsl; dic_so[rank] = dic;
    __syncthreads();

    if (h == 0) {                 // serial prefix scan over 64 sorted events
        float asl = 0.f, aic = b2[d];
        for (int j = 0; j < H_SZ; ++j) { asl += bsl_s[j]; aic += bic_s[j]; }
        seg_sl[0] = asl; seg_ic[0] = aic;
        for (int i = 0; i < H_SZ; ++i) {
            asl += dsl_so[i]; aic += dic_so[i];
            seg_sl[i + 1] = asl; seg_ic[i + 1] = aic;
        }
    }
    __syncthreads();

    BP[h * D_SZ + d] = t_so[h];
    SI[h * D_SZ + d] = make_float2(seg_sl[h], seg_ic[h]);
    if (h == 0) {
        SI[H_SZ * D_SZ + d] = make_float2(seg_sl[H_SZ], seg_ic[H_SZ]);
    }
}

// ---------------------------------------------------------------------------
// Fused main kernel. Stage 1: u[b,d] via PWL binary search (or direct MLP
// fallback). Stage 2: out = u @ Wc^T + bc with exact-f32 WMMA 16x16x4.
// ---------------------------------------------------------------------------
template <bool USE_PWL>
__global__ __launch_bounds__(256) void kan_fused_kernel(
    const float* __restrict__ x,   // [B, D]
    const float* __restrict__ W1,  // [D, H]
    const float* __restrict__ b1,  // [D, H]
    const float* __restrict__ W2,  // [D, H]
    const float* __restrict__ b2,  // [D]
    const float* __restrict__ Wc,  // [O, D]
    const float* __restrict__ bc,  // [O]
    const float* __restrict__ TBL, // d_ws tables (PWL path) or nullptr
    float* __restrict__ out)       // [B, O]
{
    extern __shared__ float smem[];
    const int tid = threadIdx.x;   // 0..255, also the feature index d
    const int b0  = blockIdx.x * BM;

    // Prefetch Wc (256KB, L2-resident, reused by every block).
    __builtin_prefetch(Wc + (size_t)tid * 256, 0, 0);

    float* u_lds;

    if (USE_PWL) {
        // LDS: [BP 64x256 f32 | SI 65x256 float2 | u BMxUPAD]
        float*        BPs = smem;
        const float2* SIs = reinterpret_cast<const float2*>(smem + BP_F);
        u_lds             = smem + TBL_F;

        // ---- Stage 199KB of tables L2 -> LDS ----
#if KAN_ASYNC_COPY
        // Dedicated async Global->LDS path (no VGPR staging, ASYNCcnt-tracked).
        // Generic LDS pointers carry the LDS byte offset in their low 32 bits.
        for (int i = tid * 4; i < TBL_F; i += 256 * 4) {
            const unsigned lds_off = (unsigned)(size_t)(smem + i);
            const float*   gp      = TBL + i;
            asm volatile("global_load_async_to_lds_b128 %0, %1, off"
                         :: "v"(lds_off), "v"(gp) : "memory");
        }
        asm volatile("s_wait_asynccnt 0x0" ::: "memory");
#else
        for (int i = tid * 4; i < TBL_F; i += 256 * 4) {
            const float4 v = *reinterpret_cast<const float4*>(TBL + i);
            *reinterpret_cast<float4*>(smem + i) = v;
        }
#endif
        __syncthreads();

        // ---- Stage 1 (PWL): 6-step branchless search + b64 pair + fma ----
        #pragma unroll 8
        for (int r = 0; r < BM; ++r) {
            const float xv = x[(size_t)(b0 + r) * D_SZ + tid];   // coalesced
            int p = 0;                                           // #breakpoints <= xv
            #pragma unroll
            for (int s = 32; s > 0; s >>= 1) {
                const float tq = BPs[(p + s - 1) * D_SZ + tid];  // conflict-free gather
                p += (tq <= xv) ? s : 0;
            }
            const float2 si = SIs[p * D_SZ + tid];               // one ds_load_b64
            u_lds[r * UPAD + tid] = fmaf(xv, si.x, si.y);
        }
    } else {
        // ---- Stage 1 (direct fallback): register-resident weights ----
        u_lds = smem;
        float w1r[H_SZ], b1r[H_SZ], w2r[H_SZ];
        {
            const float4* p1 = reinterpret_cast<const float4*>(W1 + (size_t)tid * H_SZ);
            const float4* pb = reinterpret_cast<const float4*>(b1 + (size_t)tid * H_SZ);
            const float4* p2 = reinterpret_cast<const float4*>(W2 + (size_t)tid * H_SZ);
            #pragma unroll
            for (int i = 0; i < H_SZ / 4; ++i) {
                float4 a = p1[i];
                w1r[4*i+0]=a.x; w1r[4*i+1]=a.y; w1r[4*i+2]=a.z; w1r[4*i+3]=a.w;
                float4 bb = pb[i];
                b1r[4*i+0]=bb.x; b1r[4*i+1]=bb.y; b1r[4*i+2]=bb.z; b1r[4*i+3]=bb.w;
                float4 cc = p2[i];
                w2r[4*i+0]=cc.x; w2r[4*i+1]=cc.y; w2r[4*i+2]=cc.z; w2r[4*i+3]=cc.w;
            }
        }
        const float b2s = b2[tid];
        #pragma unroll 4
        for (int r = 0; r < BM; ++r) {
            const float xv = x[(size_t)(b0 + r) * D_SZ + tid];
            float acc = b2s;
            #pragma unroll
            for (int h = 0; h < H_SZ; ++h) {
                float t = fmaf(xv, w1r[h], b1r[h]);
                t = fmaxf(t, 0.0f);
                acc = fmaf(t, w2r[h], acc);
            }
            u_lds[r * UPAD + tid] = acc;
        }
    }
    __syncthreads();

    // ---- Stage 2: out[64,256] = u[64,256] @ Wc^T + bc (exact f32 WMMA) ----
    const int lane    = tid & 31;
    const int wave    = tid >> 5;
    const int l16     = lane & 15;
    const int halfsel = lane >> 4;          // 0: K={0,1}, 1: K={2,3}
    const int n0a = (wave * 2 + 0) * 16;
    const int n0b = (wave * 2 + 1) * 16;

    v8f acc[4][2];
    #pragma unroll
    for (int m = 0; m < 4; ++m) {
        #pragma unroll
        for (int j = 0; j < 2; ++j) acc[m][j] = (v8f){0,0,0,0,0,0,0,0};
    }

    const float* wcA = Wc + (size_t)(n0a + l16) * D_SZ + 2 * halfsel;
    const float* wcB = Wc + (size_t)(n0b + l16) * D_SZ + 2 * halfsel;

    for (int k = 0; k < D_SZ; k += 4) {
        const v2f bfA = *reinterpret_cast<const v2f*>(wcA + k);
        const v2f bfB = *reinterpret_cast<const v2f*>(wcB + k);
        #pragma unroll
        for (int m = 0; m < 4; ++m) {
            const float* ap = u_lds + (m * 16 + l16) * UPAD + k + 2 * halfsel;
            const v2f af = *reinterpret_cast<const v2f*>(ap);   // ds_load_b64, conflict-free
            acc[m][0] = __builtin_amdgcn_wmma_f32_16x16x4_f32(
                false, af, false, bfA, (short)0, acc[m][0], false, false);
            acc[m][1] = __builtin_amdgcn_wmma_f32_16x16x4_f32(
                false, af, false, bfB, (short)0, acc[m][1], false, false);
        }
    }

    #pragma unroll
    for (int j = 0; j < 2; ++j) {
        const int n0  = (wave * 2 + j) * 16;
        const float bcv = bc[n0 + l16];
        #pragma unroll
        for (int m = 0; m < 4; ++m) {
            #pragma unroll
            for (int v = 0; v < 8; ++v) {
                const int row = m * 16 + v + 8 * halfsel;
                out[(size_t)(b0 + row) * O_SZ + n0 + l16] = acc[m][j][v] + bcv;
            }
        }
    }
}

extern "C" void kernel_launch(void* const* d_in, const int* in_sizes, int n_in,
                              void* d_out, int out_size, void* d_ws, size_t ws_size,
                              hipStream_t stream) {
    const float* x  = (const float*)d_in[0];
    const float* W1 = (const float*)d_in[1];
    const float* b1 = (const float*)d_in[2];
    const float* W2 = (const float*)d_in[3];
    const float* b2 = (const float*)d_in[4];
    const float* Wc = (const float*)d_in[5];
    const float* bc = (const float*)d_in[6];
    float* out = (float*)d_out;

    if (ws_size >= (size_t)TBL_F * sizeof(float)) {
        float*  BP = (float*)d_ws;
        float2* SI = (float2*)(BP + BP_F);
        kan_build_pwl<<<D_SZ, H_SZ, 0, stream>>>(W1, b1, W2, b2, BP, SI);

        const size_t shmem = (size_t)SMEM_PWL_F * sizeof(float);   // 265,216 B
        (void)hipFuncSetAttribute(reinterpret_cast<const void*>(kan_fused_kernel<true>),
                                  hipFuncAttributeMaxDynamicSharedMemorySize, (int)shmem);
        kan_fused_kernel<true><<<B_SZ / BM, 256, shmem, stream>>>(
            x, W1, b1, W2, b2, Wc, bc, BP, out);
    } else {
        const size_t shmem = (size_t)SMEM_DIR_F * sizeof(float);   // 66,560 B
        (void)hipFuncSetAttribute(reinterpret_cast<const void*>(kan_fused_kernel<false>),
                                  hipFuncAttributeMaxDynamicSharedMemorySize, (int)shmem);
        kan_fused_kernel<false><<<B_SZ / BM, 256, shmem, stream>>>(
            x, W1, b1, W2, b2, Wc, bc, nullptr, out);
    }
}